// NUFFT_24111946400473
// MI455X (gfx1250) — compile-verified
//
#include <hip/hip_runtime.h>
#include <hip/hip_bf16.h>
#include <math.h>

// ---------------------------------------------------------------------------
// NUFFT type-2 on gfx1250.
//   grid = FFT2(ifftshift(pad(img / apod)))  -> dense DFT complex GEMMs with
//   V_WMMA_F32_16X16X4_F32. Block = 8 waves covering 128(M) x 64(N); the B
//   chunk (16K x 64N x {re,im}) is staged in LDS once per block and consumed
//   via ds_load by all 8 waves; each wave register-blocks 4 N-tiles so one A
//   load feeds 16 WMMAs per K-step (~44 FLOP/B from L2 -> compute bound).
//   out = KB 6x6 gather from the 640x640 grid (L2-resident, VALU gather).
// ---------------------------------------------------------------------------

typedef __attribute__((ext_vector_type(2))) float v2f;
typedef __attribute__((ext_vector_type(8))) float v8f;

#define GSZ   640          // oversampled grid
#define HSZ   320          // image size
#define NBC   32           // B*C images
#define KPTS  40960        // non-uniform points
#define JW    6            // KB kernel width
#define KBALPHA 14.04f     // 2.34 * J
#define PI_F  3.14159265358979323846f

#define KC    16           // K-chunk staged in LDS per block iteration
#define NB    64           // N columns per block (4 N-tiles per wave)
#define MB    128          // M rows per block (8 waves x 16)
#define LDN   65           // padded LDS row length (bank-conflict free)

// f32 WMMA 16x16x4: A = 16x4 (2 VGPR/lane), B = 4x16 (2 VGPR/lane), C/D = v8f
__device__ __forceinline__ v8f wmma4(v2f a, v2f b, v8f c) {
    return __builtin_amdgcn_wmma_f32_16x16x4_f32(
        /*neg_a=*/false, a, /*neg_b=*/false, b,
        /*c_mod=*/(short)0, c, /*reuse_a=*/false, /*reuse_b=*/false);
}

// ---------------------------------------------------------------------------
// 0) DFT twiddle matrix W[u][y] = exp(-2*pi*i*u*y/G), exact phase via mod G.
// ---------------------------------------------------------------------------
__global__ void twiddle_kernel(float* __restrict__ wr, float* __restrict__ wi) {
    int i = blockIdx.x * blockDim.x + threadIdx.x;
    if (i >= GSZ * GSZ) return;
    int u = i / GSZ, y = i % GSZ;
    int m = (int)(((long long)u * (long long)y) % GSZ);   // exact in int
    float ang = -2.0f * PI_F * (float)m / (float)GSZ;
    float s, c;
    __sincosf(ang, &s, &c);
    wr[i] = c;
    wi[i] = s;
}

// ---------------------------------------------------------------------------
// 1) Deapodize + centered zero-pad + ifftshift, producing X[bc][y][x] planes.
// ---------------------------------------------------------------------------
__device__ __forceinline__ float kb_ft_dev(int idx) {   // idx in [0, 320)
    float f   = (float)(idx - HSZ / 2) / (float)GSZ;    // |f| <= 0.25
    float pjf = PI_F * (float)JW * f;
    float z   = sqrtf(KBALPHA * KBALPHA - pjf * pjf);   // always real here
    float sh  = 0.5f * (expf(z) - expf(-z));            // sinh(z)
    return (float)JW * sh / z;
}

__global__ void prep_kernel(const float* __restrict__ imr,
                            const float* __restrict__ imi,
                            float* __restrict__ xr, float* __restrict__ xi) {
    long long i = (long long)blockIdx.x * blockDim.x + threadIdx.x;
    const long long N = (long long)NBC * GSZ * GSZ;
    if (i >= N) return;
    int gx = (int)(i % GSZ);
    int gy = (int)((i / GSZ) % GSZ);
    int bc = (int)(i / ((long long)GSZ * GSZ));
    // ifftshift (even G: roll by G/2) of the centered-padded image
    int sy = (gy + GSZ / 2) % GSZ;
    int sx = (gx + GSZ / 2) % GSZ;
    const int p0 = (GSZ - HSZ) / 2;   // 160
    float vr = 0.0f, vi = 0.0f;
    if (sy >= p0 && sy < p0 + HSZ && sx >= p0 && sx < p0 + HSZ) {
        int iy = sy - p0, ix = sx - p0;
        long long src = ((long long)bc * HSZ + iy) * HSZ + ix;
        float d = kb_ft_dev(iy) * kb_ft_dev(ix);
        vr = imr[src] / d;
        vi = imi[src] / d;
    }
    xr[i] = vr;
    xi[i] = vi;
}

// ---------------------------------------------------------------------------
// 2) Column DFT: Y[u][x] = sum_y W[u][y] * X[y][x]  (complex, per image).
//    Block: img fixed, 8 waves = 8 M-tiles (rows of W), shared 64 N columns.
//    B chunk X[ks..ks+15][n0..n0+63] {re,im} staged in LDS; A (twiddle rows)
//    streamed from L2 as b64 pairs per lane.
//    WMMA f32 16x16x4 lane layout (ISA 7.12.2):
//      A: lanes 0-15 hold K=ks,ks+1; lanes 16-31 hold K=ks+2,ks+3
//      B: VGPR0 = K lo-half / K+2 hi-half, VGPR1 = K+1 / K+3
//      D: VGPR r -> row r (lanes 0-15) / row r+8 (lanes 16-31)
// ---------------------------------------------------------------------------
__global__ void dft_col_kernel(const float* __restrict__ wr,
                               const float* __restrict__ wi,
                               const float* __restrict__ xr,
                               const float* __restrict__ xi,
                               float* __restrict__ yr, float* __restrict__ yi) {
    __shared__ float Bs[2][KC][LDN];          // {re,im} x K x N (padded)

    const int TPM = GSZ / MB;                 // 5 M-blocks
    const int TPN = GSZ / NB;                 // 10 N-blocks
    int bid  = blockIdx.x;
    int img  = bid / (TPM * TPN);
    int rem  = bid % (TPM * TPN);
    int u0   = (rem / TPN) * MB + (threadIdx.x >> 5) * 16;  // this wave's M-tile
    int n0   = (rem % TPN) * NB;
    int lane = threadIdx.x & 31;
    int half = lane >> 4;
    int lm   = lane & 15;

    const float* Xr = xr + (long long)img * GSZ * GSZ;
    const float* Xi = xi + (long long)img * GSZ * GSZ;

    v8f cr0 = {}, cr1 = {}, cr2 = {}, cr3 = {};
    v8f ci0 = {}, ci1 = {}, ci2 = {}, ci3 = {};

    for (int ks = 0; ks < GSZ; ks += KC) {
        __syncthreads();
        // stage B chunk: thread-minor = n (row-contiguous, coalesced)
        for (int e = threadIdx.x; e < 2 * KC * NB; e += 256) {
            int plane = e >> 10;              // KC*NB = 1024
            int r2    = e & 1023;
            int k     = r2 >> 6;
            int n     = r2 & 63;
            const float* src = plane ? Xi : Xr;
            Bs[plane][k][n] = src[(ks + k) * GSZ + n0 + n];
        }
        __syncthreads();

#pragma unroll
        for (int kk = 0; kk < KC; kk += 4) {
            int ka = kk + 2 * half;
            int arow = (u0 + lm) * GSZ + ks + ka;       // twiddle [u][y]
            v2f ar, aim, an;
            ar.x  = wr[arow];  ar.y  = wr[arow + 1];
            aim.x = wi[arow];  aim.y = wi[arow + 1];
            an.x  = -aim.x;    an.y  = -aim.y;          // f32 WMMA has no A-neg

#define COL_TILE(J, CR, CI)                                            \
            {                                                          \
                v2f br, bim;                                           \
                int n = (J) * 16 + lm;                                 \
                br.x  = Bs[0][ka][n];     br.y  = Bs[0][ka + 1][n];    \
                bim.x = Bs[1][ka][n];     bim.y = Bs[1][ka + 1][n];    \
                CR = wmma4(ar,  br,  CR);                              \
                CR = wmma4(an,  bim, CR);                              \
                CI = wmma4(ar,  bim, CI);                              \
                CI = wmma4(aim, br,  CI);                              \
            }
            COL_TILE(0, cr0, ci0)
            COL_TILE(1, cr1, ci1)
            COL_TILE(2, cr2, ci2)
            COL_TILE(3, cr3, ci3)
#undef COL_TILE
        }
    }

    long long ob = (long long)img * GSZ * GSZ;
#define COL_STORE(J, CR, CI)                                           \
    {                                                                  \
        _Pragma("unroll")                                              \
        for (int r = 0; r < 8; ++r) {                                  \
            int row = u0 + r + 8 * half;                               \
            long long o = ob + (long long)row * GSZ + n0 + (J)*16 + lm;\
            yr[o] = CR[r];                                             \
            yi[o] = CI[r];                                             \
        }                                                              \
    }
    COL_STORE(0, cr0, ci0)
    COL_STORE(1, cr1, ci1)
    COL_STORE(2, cr2, ci2)
    COL_STORE(3, cr3, ci3)
#undef COL_STORE
}

// ---------------------------------------------------------------------------
// 3) Row DFT: Z[u][v] = sum_x Y[u][x] * W[v][x].  A = data rows (streamed),
//    B[k][n] = W[v0+n][k] (twiddle transposed-read) staged in LDS with
//    thread-minor = k so each lane group reads 16 contiguous floats.
// ---------------------------------------------------------------------------
__global__ void dft_row_kernel(const float* __restrict__ wr,
                               const float* __restrict__ wi,
                               const float* __restrict__ yr,
                               const float* __restrict__ yi,
                               float* __restrict__ zr, float* __restrict__ zi) {
    __shared__ float Bs[2][KC][LDN];

    const int TPM = GSZ / MB;
    const int TPN = GSZ / NB;
    int bid  = blockIdx.x;
    int img  = bid / (TPM * TPN);
    int rem  = bid % (TPM * TPN);
    int u0   = (rem / TPN) * MB + (threadIdx.x >> 5) * 16;
    int v0   = (rem % TPN) * NB;
    int lane = threadIdx.x & 31;
    int half = lane >> 4;
    int lm   = lane & 15;

    const float* Yr = yr + (long long)img * GSZ * GSZ;
    const float* Yi = yi + (long long)img * GSZ * GSZ;

    v8f cr0 = {}, cr1 = {}, cr2 = {}, cr3 = {};
    v8f ci0 = {}, ci1 = {}, ci2 = {}, ci3 = {};

    for (int ks = 0; ks < GSZ; ks += KC) {
        __syncthreads();
        // stage B chunk: B[k][n] = W[v0+n][ks+k]; thread-minor = k keeps
        // 16-float contiguous runs per twiddle row (padded LDS row avoids
        // the stride-64 bank conflict on the store side)
        for (int e = threadIdx.x; e < 2 * KC * NB; e += 256) {
            int plane = e >> 10;
            int r2    = e & 1023;
            int n     = r2 >> 4;
            int k     = r2 & 15;
            const float* src = plane ? wi : wr;
            Bs[plane][k][n] = src[(v0 + n) * GSZ + ks + k];
        }
        __syncthreads();

#pragma unroll
        for (int kk = 0; kk < KC; kk += 4) {
            int ka = kk + 2 * half;
            int arow = (u0 + lm) * GSZ + ks + ka;       // Y row-major [u][x]
            v2f ar, aim, an;
            ar.x  = Yr[arow];  ar.y  = Yr[arow + 1];
            aim.x = Yi[arow];  aim.y = Yi[arow + 1];
            an.x  = -aim.x;    an.y  = -aim.y;

#define ROW_TILE(J, CR, CI)                                            \
            {                                                          \
                v2f br, bim;                                           \
                int n = (J) * 16 + lm;                                 \
                br.x  = Bs[0][ka][n];     br.y  = Bs[0][ka + 1][n];    \
                bim.x = Bs[1][ka][n];     bim.y = Bs[1][ka + 1][n];    \
                CR = wmma4(ar,  br,  CR);                              \
                CR = wmma4(an,  bim, CR);                              \
                CI = wmma4(ar,  bim, CI);                              \
                CI = wmma4(aim, br,  CI);                              \
            }
            ROW_TILE(0, cr0, ci0)
            ROW_TILE(1, cr1, ci1)
            ROW_TILE(2, cr2, ci2)
            ROW_TILE(3, cr3, ci3)
#undef ROW_TILE
        }
    }

    long long ob = (long long)img * GSZ * GSZ;
#define ROW_STORE(J, CR, CI)                                           \
    {                                                                  \
        _Pragma("unroll")                                              \
        for (int r = 0; r < 8; ++r) {                                  \
            int row = u0 + r + 8 * half;                               \
            long long o = ob + (long long)row * GSZ + v0 + (J)*16 + lm;\
            zr[o] = CR[r];                                             \
            zi[o] = CI[r];                                             \
        }                                                              \
    }
    ROW_STORE(0, cr0, ci0)
    ROW_STORE(1, cr1, ci1)
    ROW_STORE(2, cr2, ci2)
    ROW_STORE(3, cr3, ci3)
#undef ROW_STORE
}

// ---------------------------------------------------------------------------
// 4) KB 6x6 interpolation gather (grid is L2-resident; bandwidth-bound).
// ---------------------------------------------------------------------------
__device__ __forceinline__ float bessel_i0(float x) {
    float ax = fabsf(x);
    if (ax < 3.75f) {
        float t = x / 3.75f; t *= t;
        return 1.0f + t*(3.5156229f + t*(3.0899424f + t*(1.2067492f +
                     t*(0.2659732f + t*(0.0360768f + t*0.0045813f)))));
    }
    float t = 3.75f / ax;
    return (expf(ax) * __frsqrt_rn(ax)) *
           (0.39894228f + t*(0.01328592f + t*(0.00225319f + t*(-0.00157565f +
            t*(0.00916281f + t*(-0.02057706f + t*(0.02635537f +
            t*(-0.01647633f + t*0.00392377f))))))));
}

__device__ __forceinline__ float kb_weight(float u) {
    float r = 2.0f * u / (float)JW;
    float arg = 1.0f - r * r;
    if (arg < 0.0f) return 0.0f;
    return bessel_i0(KBALPHA * sqrtf(arg));
}

__global__ void interp_kernel(const float* __restrict__ traj,
                              const float* __restrict__ gr,
                              const float* __restrict__ gi,
                              float* __restrict__ out) {
    int t = blockIdx.x * blockDim.x + threadIdx.x;
    if (t >= NBC * KPTS) return;
    int k  = t % KPTS;
    int bc = t / KPTS;
    const float scale = (float)GSZ / (2.0f * PI_F);
    float t0 = traj[k] * scale;           // axis 0 (grid rows)
    float t1 = traj[KPTS + k] * scale;    // axis 1 (grid cols)
    float k00 = floorf(t0), k01 = floorf(t1);
    float w0[JW], w1[JW];
    int   i0a[JW], i1a[JW];
#pragma unroll
    for (int j = 0; j < JW; ++j) {
        float off = (float)j - (float)(JW / 2 - 1);   // -2..3
        float id0 = k00 + off, id1 = k01 + off;
        w0[j] = kb_weight(t0 - id0);
        w1[j] = kb_weight(t1 - id1);
        int m0 = (int)id0 % GSZ; if (m0 < 0) m0 += GSZ;
        int m1 = (int)id1 % GSZ; if (m1 < 0) m1 += GSZ;
        i0a[j] = m0; i1a[j] = m1;
    }
    const float* Gr = gr + (long long)bc * GSZ * GSZ;
    const float* Gi = gi + (long long)bc * GSZ * GSZ;
    float sr = 0.0f, si = 0.0f;
#pragma unroll
    for (int a = 0; a < JW; ++a) {
        int rowoff = i0a[a] * GSZ;
        float wa = w0[a];
#pragma unroll
        for (int b = 0; b < JW; ++b) {
            float w = wa * w1[b];
            int lin = rowoff + i1a[b];
            sr = fmaf(Gr[lin], w, sr);
            si = fmaf(Gi[lin], w, si);
        }
    }
    out[(long long)bc * KPTS + k] = sr;                              // real plane
    out[(long long)NBC * KPTS + (long long)bc * KPTS + k] = si;      // imag plane
}

// ---------------------------------------------------------------------------
extern "C" void kernel_launch(void* const* d_in, const int* in_sizes, int n_in,
                              void* d_out, int out_size, void* d_ws, size_t ws_size,
                              hipStream_t stream) {
    const float* imr  = (const float*)d_in[0];   // [4,8,320,320]
    const float* imi  = (const float*)d_in[1];   // [4,8,320,320]
    const float* traj = (const float*)d_in[2];   // [2,40960]

    float* ws = (float*)d_ws;
    const long long PLANE = (long long)GSZ * GSZ;     //   409,600 floats
    const long long IMGS  = (long long)NBC * PLANE;   // 13,107,200 floats
    float* wr = ws;                 // twiddle real
    float* wi = wr + PLANE;         // twiddle imag
    float* xr = wi + PLANE;         // ping planes (prep out / final grid)
    float* xi = xr + IMGS;
    float* yr = xi + IMGS;          // pong planes (after column DFT)
    float* yi = yr + IMGS;

    twiddle_kernel<<<(GSZ * GSZ + 255) / 256, 256, 0, stream>>>(wr, wi);

    prep_kernel<<<(int)((IMGS + 255) / 256), 256, 0, stream>>>(imr, imi, xr, xi);

    // 32 images x (640/128) M-blocks x (640/64) N-blocks = 1600 blocks/pass
    const int blocks = NBC * (GSZ / MB) * (GSZ / NB);
    dft_col_kernel<<<blocks, 256, 0, stream>>>(wr, wi, xr, xi, yr, yi);
    dft_row_kernel<<<blocks, 256, 0, stream>>>(wr, wi, yr, yi, xr, xi);

    interp_kernel<<<(NBC * KPTS + 255) / 256, 256, 0, stream>>>(traj, xr, xi,
                                                                (float*)d_out);
}